// tconv4a_56813827391719
// MI455X (gfx1250) — compile-verified
//
#include <hip/hip_runtime.h>
#include <hip/hip_bf16.h>

// ---------------------------------------------------------------------------
// Tucker-2 tensor-conv ResNet forward for MI455X (gfx1250, wave32, WMMA).
// Core math per conv offset (i,j):  OUT(17x17) = W1[ij] @ X(17x17) @ W2[ij]
// batched over spatial pixels. Implemented as two K=17 GEMM stages using
// V_WMMA_F32_16X16X4_F32 (fp32 matrix core), K padded 17->20 (5 wmma steps),
// M=17 padded to two 16-row tiles. One workgroup = 4x4 output pixels (N=16).
// LDS B-operands stored K-pair-interleaved (float2) so each B fragment is one
// ds_load_b64; stage-A output rows clamped to a dummy LDS row to avoid
// divergent store ladders around the WMMA pipeline.
// ---------------------------------------------------------------------------

typedef float v2f __attribute__((ext_vector_type(2)));
typedef float v8f __attribute__((ext_vector_type(8)));

#define NW 8  // waves per workgroup (256 threads, wave32)

__global__ __launch_bounds__(256) void tucker_tconv(
    const float* __restrict__ X, const float* __restrict__ W1,
    const float* __restrict__ W2, const float* __restrict__ bias,
    float* __restrict__ OUT,
    int Hin, int Win, int Hout, int Wout, int stride, int ksize, int pad)
{
    __shared__ float W1p[32 * 20];   // stage-A A-matrix: rows p (pad 32), cols a (pad 20)
    __shared__ float W2Tp[32 * 20];  // stage-B A-matrix: rows q (pad 32), cols b (pad 20)
    __shared__ v2f Xt2[17 * 10 * 16];   // X tile  [b][k2=a/2][n] K-pair interleaved
    __shared__ v2f T1b[18 * 10 * 16];   // T1 tile [p(+dummy 17)][k2=b/2][n]
    float* const Xtf = (float*)Xt2;
    float* const T1f = (float*)T1b;

    const int tid  = threadIdx.x;
    const int lane = tid & 31;
    const int wave = tid >> 5;
    const int ln16 = lane & 15;
    const int hi   = (lane >> 4) & 1;   // lanes 16..31 handle K+2 / M+8 halves
    const int noff = ksize * ksize;

    // tile decode: 4x4 output pixels per workgroup
    const int tpw = Wout >> 2;
    const int tph = Hout >> 2;
    const int tiles_per_img = tpw * tph;
    const int bimg = blockIdx.x / tiles_per_img;
    const int tt   = blockIdx.x % tiles_per_img;
    const int oy0  = (tt / tpw) * 4;
    const int ox0  = (tt % tpw) * 4;
    const int oy   = oy0 + (ln16 >> 2);   // this lane's output pixel (D column n)
    const int ox   = ox0 + (ln16 & 3);

    // zero the K=17..19 pad slots once (gather / stage-A never write them):
    // for each (row, n): Xt2[row][8].y, Xt2[row][9].x, Xt2[row][9].y
    for (int i = tid; i < 17 * 16 * 3; i += 256) {
        int row = i / 48, r = i % 48, which = r / 16, n = r % 16;
        int fi = (which == 0) ? (((row * 10 + 8) * 16 + n) * 2 + 1)
                              : (((row * 10 + 9) * 16 + n) * 2 + (which - 1));
        Xtf[fi] = 0.f;
        T1f[fi] = 0.f;  // same pad pattern for rows p=0..16 of T1
    }
    __syncthreads();

    v8f accB[5];
#pragma unroll
    for (int s = 0; s < 5; ++s) accB[s] = {};

    for (int off = 0; off < noff; ++off) {
        const int di = off / ksize - pad;
        const int dj = off % ksize - pad;

        // load this offset's weights, zero-padded
        for (int i = tid; i < 640; i += 256) {
            int m = i / 20, k = i % 20;
            float v1 = 0.f, v2 = 0.f;
            if (m < 17 && k < 17) {
                v1 = W1[(off * 17 + m) * 17 + k];  // W1[ij][p=m][a=k]
                v2 = W2[(off * 17 + k) * 17 + m];  // W2T[q=m][b=k] = W2[ij][b][q]
            }
            W1p[i]  = v1;
            W2Tp[i] = v2;
        }
        // gather X tile (shift di,dj; stride; zero-fill OOB)
        for (int i = tid; i < 17 * 17 * 16; i += 256) {
            int a = i / 272, r = i % 272, b = r / 16, n = r % 16;
            int iy = (oy0 + (n >> 2)) * stride + di;
            int ix = (ox0 + (n & 3)) * stride + dj;
            float v = 0.f;
            if (iy >= 0 && iy < Hin && ix >= 0 && ix < Win)
                v = X[(((size_t)bimg * 289 + a * 17 + b) * Hin + iy) * Win + ix];
            Xtf[((b * 10 + (a >> 1)) * 16 + n) * 2 + (a & 1)] = v;
        }
        __syncthreads();

        // ---- stage A: T1[p,b,n] = sum_a W1[p,a] * X[a,b,n] (34 jobs / 8 waves)
#pragma unroll
        for (int s = 0; s < 5; ++s) {
            int job = wave + s * NW;
            if (job < 34) {
                int b = job % 17, mt = job / 17;
                v8f acc = {};
#pragma unroll
                for (int kk = 0; kk < 5; ++kk) {
                    int kb = kk * 4 + 2 * hi;          // even, per-half K base
                    int m  = mt * 16 + ln16;
                    v2f af = *(const v2f*)&W1p[m * 20 + kb];
                    v2f bf = Xt2[(b * 10 + kk * 2 + hi) * 16 + ln16];
                    acc = __builtin_amdgcn_wmma_f32_16x16x4_f32(
                        false, af, false, bf, (short)0, acc, false, false);
                }
#pragma unroll
                for (int r = 0; r < 8; ++r) {
                    int row  = mt * 16 + r + hi * 8;   // D: vgpr r -> M=r / r+8
                    int rowc = row < 17 ? row : 17;    // clamp to dummy row, no branch
                    T1f[((rowc * 10 + (b >> 1)) * 16 + ln16) * 2 + (b & 1)] = acc[r];
                }
            }
        }
        __syncthreads();

        // ---- stage B: OUT[p,q,n] += sum_b W2T[q,b] * T1[p,b,n]
#pragma unroll
        for (int s = 0; s < 5; ++s) {
            int job = wave + s * NW;
            if (job < 34) {
                int p = job % 17, qt = job / 17;
                v8f acc = accB[s];
#pragma unroll
                for (int kk = 0; kk < 5; ++kk) {
                    int kb = kk * 4 + 2 * hi;
                    int q  = qt * 16 + ln16;
                    v2f af = *(const v2f*)&W2Tp[q * 20 + kb];
                    v2f bf = T1b[(p * 10 + kk * 2 + hi) * 16 + ln16];
                    acc = __builtin_amdgcn_wmma_f32_16x16x4_f32(
                        false, af, false, bf, (short)0, acc, false, false);
                }
                accB[s] = acc;
            }
        }
        __syncthreads();
    }

    // writeout: channel c = p*17+q, pixel (oy,ox)
#pragma unroll
    for (int s = 0; s < 5; ++s) {
        int job = wave + s * NW;
        if (job < 34) {
            int p = job % 17, qt = job / 17;
#pragma unroll
            for (int r = 0; r < 8; ++r) {
                int q = qt * 16 + r + hi * 8;
                if (q < 17) {
                    int c = p * 17 + q;
                    OUT[(((size_t)bimg * 289 + c) * Hout + oy) * Wout + ox] =
                        accB[s][r] + bias[c];
                }
            }
        }
    }
}

// features conv: 3->289 ch, 3x3 s2 pad1, dense (Cin tiny, VALU is fine)
__global__ __launch_bounds__(256) void feat_conv(
    const float* __restrict__ x, const float* __restrict__ W1,
    const float* __restrict__ W2, const float* __restrict__ bias,
    float* __restrict__ out)
{
    __shared__ float w1[9 * 17 * 3];  // [ij][p][a]
    __shared__ float w2[9 * 17];      // [ij][q]
    int tid = threadIdx.x;
    for (int i = tid; i < 9 * 17 * 3; i += 256) w1[i] = W1[i];
    for (int i = tid; i < 9 * 17; i += 256) w2[i] = W2[i];
    __syncthreads();
    const int HO = 80, WO = 80, HI = 160, WI = 160;
    size_t total = (size_t)32 * 289 * HO * WO;
    for (size_t idx = (size_t)blockIdx.x * 256 + tid; idx < total;
         idx += (size_t)gridDim.x * 256) {
        int ox = idx % WO;
        int oy = (idx / WO) % HO;
        int c  = (idx / ((size_t)WO * HO)) % 289;
        int bi = idx / ((size_t)WO * HO * 289);
        int p = c / 17, q = c % 17;
        float acc = bias[c];
        for (int i = 0; i < 3; ++i) {
            int iy = oy * 2 - 1 + i;
            if (iy < 0 || iy >= HI) continue;
            for (int j = 0; j < 3; ++j) {
                int ix = ox * 2 - 1 + j;
                if (ix < 0 || ix >= WI) continue;
                int ij = i * 3 + j;
                float s = 0.f;
                for (int a = 0; a < 3; ++a)
                    s += w1[(ij * 17 + p) * 3 + a] *
                         x[((size_t)(bi * 3 + a) * HI + iy) * WI + ix];
                acc += s * w2[ij * 17 + q];
            }
        }
        out[idx] = acc;
    }
}

__global__ void zero_f(float* p, int n) {
    int i = blockIdx.x * 256 + threadIdx.x;
    if (i < n) p[i] = 0.f;
}

// per-(channel,batch) partial sums -> atomic into st[2c], st[2c+1]
__global__ __launch_bounds__(256) void bn_reduce(
    const float* __restrict__ x, float* __restrict__ st, int HW)
{
    int c = blockIdx.x, bi = blockIdx.y;
    const float* p = x + ((size_t)bi * 289 + c) * HW;
    float s = 0.f, s2 = 0.f;
    for (int i = threadIdx.x; i < HW; i += 256) {
        float v = p[i];
        s += v; s2 += v * v;
    }
    __shared__ float ls[256], ls2[256];
    ls[threadIdx.x] = s; ls2[threadIdx.x] = s2;
    __syncthreads();
    for (int off = 128; off > 0; off >>= 1) {
        if (threadIdx.x < off) {
            ls[threadIdx.x]  += ls[threadIdx.x + off];
            ls2[threadIdx.x] += ls2[threadIdx.x + off];
        }
        __syncthreads();
    }
    if (threadIdx.x == 0) {
        atomicAdd(&st[2 * c], ls[0]);
        atomicAdd(&st[2 * c + 1], ls2[0]);
    }
}

// y = gamma*(x-m)*rsqrt(v+eps)+beta  (+ optional PReLU); in-place safe
__global__ __launch_bounds__(256) void bn_apply(
    float* __restrict__ y, const float* __restrict__ xin,
    const float* __restrict__ st, const float* __restrict__ g,
    const float* __restrict__ bt, const float* __restrict__ alpha,
    int use_prelu, int HW, float invN)
{
    size_t total = (size_t)32 * 289 * HW;
    float a = use_prelu ? alpha[0] : 0.f;
    for (size_t idx = (size_t)blockIdx.x * 256 + threadIdx.x; idx < total;
         idx += (size_t)gridDim.x * 256) {
        int c = (idx / HW) % 289;
        float m = st[2 * c] * invN;
        float v = st[2 * c + 1] * invN - m * m;
        float r = rsqrtf(v + 1e-5f);
        float val = g[c] * (xin[idx] - m) * r + bt[c];
        if (use_prelu && val < 0.f) val *= a;
        y[idx] = val;
    }
}

// out = bn(y3) + bn(idn)   (residual tail, no activation)
__global__ __launch_bounds__(256) void bn2_add(
    const float* __restrict__ y3, const float* __restrict__ idn,
    const float* __restrict__ stY, const float* __restrict__ stI,
    const float* __restrict__ gy, const float* __restrict__ bty,
    const float* __restrict__ gi, const float* __restrict__ bti,
    float* __restrict__ out, int HW, float invN)
{
    size_t total = (size_t)32 * 289 * HW;
    for (size_t idx = (size_t)blockIdx.x * 256 + threadIdx.x; idx < total;
         idx += (size_t)gridDim.x * 256) {
        int c = (idx / HW) % 289;
        float my = stY[2 * c] * invN;
        float vy = stY[2 * c + 1] * invN - my * my;
        float mi = stI[2 * c] * invN;
        float vi = stI[2 * c + 1] * invN - mi * mi;
        float ry = rsqrtf(vy + 1e-5f);
        float ri = rsqrtf(vi + 1e-5f);
        out[idx] = gy[c] * (y3[idx] - my) * ry + bty[c] +
                   gi[c] * (idn[idx] - mi) * ri + bti[c];
    }
}

extern "C" void kernel_launch(void* const* d_in, const int* in_sizes, int n_in,
                              void* d_out, int out_size, void* d_ws, size_t ws_size,
                              hipStream_t stream) {
    const float* x     = (const float*)d_in[0];
    const float* fW1   = (const float*)d_in[1];
    const float* fW2   = (const float*)d_in[2];
    const float* fb    = (const float*)d_in[3];
    const float* c11W1 = (const float*)d_in[4];
    const float* c11W2 = (const float*)d_in[5];
    const float* c11b  = (const float*)d_in[6];
    const float* c33W1 = (const float*)d_in[7];
    const float* c33W2 = (const float*)d_in[8];
    const float* c33b  = (const float*)d_in[9];
    const float* bng   = (const float*)d_in[10];
    const float* bnbt  = (const float*)d_in[11];
    const float* pa    = (const float*)d_in[12];
    float* out = (float*)d_out;

    const size_t SZ80 = 32ull * 289 * 80 * 80;
    const size_t SZ40 = 32ull * 289 * 40 * 40;
    float* ws  = (float*)d_ws;
    float* h   = ws;
    float* t0  = h + SZ80;
    float* t1  = t0 + SZ80;
    float* idn = t1 + SZ40;
    float* st  = idn + SZ40;  // 4 stat slots of 2*289 floats

    // features: dense conv -> BN -> PReLU
    zero_f<<<3, 256, 0, stream>>>(st, 578);
    feat_conv<<<4096, 256, 0, stream>>>(x, fW1, fW2, fb, h);
    bn_reduce<<<dim3(289, 32), 256, 0, stream>>>(h, st, 6400);
    {
        size_t total = 32ull * 289 * 6400;
        bn_apply<<<(unsigned)((total + 255) / 256), 256, 0, stream>>>(
            h, h, st, bng, bnbt, pa, 1, 6400, 1.f / (32.f * 6400.f));
    }

    for (int k = 0; k < 9; ++k) {
        int Sin  = (k == 0) ? 80 : ((k == 1) ? 40 : 20);
        int Sout = (k < 2) ? Sin / 2 : Sin;
        int s    = (k < 2) ? 2 : 1;
        int HWin = Sin * Sin, HWout = Sout * Sout;
        float invNin  = 1.f / (32.f * HWin);
        float invNout = 1.f / (32.f * HWout);
        int j = 10 + 3 * k;
        float* st0 = st;
        float* st1 = st + 578;
        float* st2 = st + 2 * 578;
        float* st3 = st + 3 * 578;
        int gin  = 32 * (Sin / 4) * (Sin / 4);
        int gout = 32 * (Sout / 4) * (Sout / 4);
        unsigned gein  = (unsigned)((32ull * 289 * HWin + 255) / 256);
        unsigned geout = (unsigned)((32ull * 289 * HWout + 255) / 256);

        zero_f<<<10, 256, 0, stream>>>(st, 4 * 578);

        // y1 = prelu(bn(tconv1x1(h)))
        tucker_tconv<<<gin, 256, 0, stream>>>(
            h, c11W1 + (size_t)(9 + 2 * k) * 289, c11W2 + (size_t)(9 + 2 * k) * 289,
            c11b + (size_t)(9 + 2 * k) * 289, t0, Sin, Sin, Sin, Sin, 1, 1, 0);
        bn_reduce<<<dim3(289, 32), 256, 0, stream>>>(t0, st0, HWin);
        bn_apply<<<gein, 256, 0, stream>>>(t0, t0, st0, bng + (size_t)j * 289,
                                           bnbt + (size_t)j * 289, pa + (1 + 2 * k),
                                           1, HWin, invNin);
        // y2 = prelu(bn(tconv3x3(y1, stride s)))
        tucker_tconv<<<gout, 256, 0, stream>>>(
            t0, c33W1 + (size_t)k * 2601, c33W2 + (size_t)k * 2601,
            c33b + (size_t)k * 289, t1, Sin, Sin, Sout, Sout, s, 3, 1);
        bn_reduce<<<dim3(289, 32), 256, 0, stream>>>(t1, st1, HWout);
        bn_apply<<<geout, 256, 0, stream>>>(t1, t1, st1, bng + (size_t)(j + 1) * 289,
                                            bnbt + (size_t)(j + 1) * 289,
                                            pa + (2 + 2 * k), 1, HWout, invNout);
        // y3 = tconv1x1(y2)
        tucker_tconv<<<gout, 256, 0, stream>>>(
            t1, c11W1 + (size_t)(10 + 2 * k) * 289, c11W2 + (size_t)(10 + 2 * k) * 289,
            c11b + (size_t)(10 + 2 * k) * 289, t0, Sout, Sout, Sout, Sout, 1, 1, 0);
        bn_reduce<<<dim3(289, 32), 256, 0, stream>>>(t0, st2, HWout);
        // idn = tconv1x1(h, stride s)
        tucker_tconv<<<gout, 256, 0, stream>>>(
            h, c11W1 + (size_t)k * 289, c11W2 + (size_t)k * 289,
            c11b + (size_t)k * 289, idn, Sin, Sin, Sout, Sout, s, 1, 0);
        bn_reduce<<<dim3(289, 32), 256, 0, stream>>>(idn, st3, HWout);
        // h = bn(y3) + bn(idn)
        float* dst = (k == 8) ? out : h;
        bn2_add<<<geout, 256, 0, stream>>>(t0, idn, st2, st3,
                                           bng + (size_t)(j + 2) * 289,
                                           bnbt + (size_t)(j + 2) * 289,
                                           bng + (size_t)(1 + k) * 289,
                                           bnbt + (size_t)(1 + k) * 289,
                                           dst, HWout, invNout);
    }
}